// Network_2216203125573
// MI455X (gfx1250) — compile-verified
//
#include <hip/hip_runtime.h>
#include <hip/hip_bf16.h>

typedef float v2f __attribute__((ext_vector_type(2)));
typedef float v8f __attribute__((ext_vector_type(8)));

#define N_ATOMS   8000
#define N_MEMBERS 2000
#define D_IN      25
#define DPAD      32
#define KC        320   // K chunk per block iteration
#define KW        40    // K slice per wave within a chunk (8 waves * 40 = 320)
#define ASTRIDE   324   // 324 % 64 == 4 -> conflict-free b64 A-fragment reads, rows 16B aligned

__device__ __forceinline__ v8f wmma_f32(v2f a, v2f b, v8f c) {
  // V_WMMA_F32_16X16X4_F32: D = A(16x4,f32) * B(4x16,f32) + C(16x16,f32)
  return __builtin_amdgcn_wmma_f32_16x16x4_f32(
      /*neg_a=*/false, a, /*neg_b=*/false, b,
      /*c_mod=*/(short)0, c, /*reuse_a=*/false, /*reuse_b=*/false);
}

// CDNA5 async direct-to-LDS copy (ASYNCcnt-tracked, no VGPR round trip).
// GV mode: each lane copies 16B from global address g to LDS offset lds_off.
__device__ __forceinline__ void async_b128(unsigned lds_off, const float* g) {
  asm volatile("global_load_async_to_lds_b128 %0, %1, off"
               :: "v"(lds_off), "v"(g)
               : "memory");
}
__device__ __forceinline__ void wait_async0() {
  asm volatile("s_wait_asynccnt 0x0" ::: "memory");
}
__device__ __forceinline__ unsigned lds_off_of(const void* p) {
  return (unsigned)(uintptr_t)p;   // generic addr of LDS object: low 32 bits = LDS byte offset
}

// ---------------------------------------------------------------------------
// h[a][d] = embed[atoms[a]][d] for d<25, 0 for pad columns 25..31
// ---------------------------------------------------------------------------
__global__ __launch_bounds__(256) void k_gather(const int* __restrict__ atoms,
                                                const float* __restrict__ embed,
                                                float* __restrict__ h) {
  const int t = blockIdx.x * 256 + threadIdx.x;
  if (t >= N_ATOMS * DPAD) return;
  const int a = t / DPAD, d = t % DPAD;
  h[t] = (d < D_IN) ? embed[atoms[a] * D_IN + d] : 0.0f;
}

// ---------------------------------------------------------------------------
// C[M x 32] = A[M x K] @ B[K x 32].  A is the HBM stream: double-buffered
// async-to-LDS staging overlapped with WMMA on the other buffer.
// One block = 16-row strip; 8 waves split K (40 each per 320-chunk).
// ---------------------------------------------------------------------------
__global__ __launch_bounds__(256) void k_gemm_wmma(const float* __restrict__ A,
                                                   const float* __restrict__ B,
                                                   float* __restrict__ C,
                                                   int K) {
  __shared__ __align__(16) float As[2][16][ASTRIDE];
  __shared__ __align__(16) float Bs[2][KC][40];   // stride 40: khalf groups on disjoint banks
  __shared__ float red[8][256];

  const int tid   = threadIdx.x;
  const int wave  = tid >> 5;
  const int lane  = tid & 31;
  const int mrow  = lane & 15;
  const int khalf = (lane >> 4) << 1;             // lanes 16..31 own K+2,K+3
  const long rowBase = (long)blockIdx.x * 16;
  const int  nch  = K / KC;

  // ---- async stage of one K-chunk into buffer `nb` ----
  auto stage = [&](int kc, int nb) {
    // A tile: 16 x 320 floats = 1280 x 16B segments
    for (int s = tid; s < 16 * (KC / 4); s += 256) {
      const int r  = s / (KC / 4);
      const int c4 = s % (KC / 4);
      async_b128(lds_off_of(&As[nb][r][c4 * 4]),
                 A + (rowBase + r) * (long)K + kc + c4 * 4);
    }
    // B tile: 320 x 32 floats = 2560 x 16B segments
    for (int s = tid; s < KC * (DPAD / 4); s += 256) {
      const int r  = s / (DPAD / 4);
      const int c4 = s % (DPAD / 4);
      async_b128(lds_off_of(&Bs[nb][r][c4 * 4]),
                 B + (long)(kc + r) * DPAD + c4 * 4);
    }
  };

  v8f c0 = {}; v8f c1 = {};

  stage(0, 0);
  wait_async0();
  __syncthreads();

  for (int c = 0; c < nch; ++c) {
    const int buf = c & 1;
    if (c + 1 < nch) stage((c + 1) * KC, buf ^ 1);   // background fill of other buffer

    const int kw0 = wave * KW;
    #pragma unroll
    for (int ks = 0; ks < KW; ks += 4) {
      const int kk = kw0 + ks + khalf;
      v2f a;  a.x  = As[buf][mrow][kk];      a.y  = As[buf][mrow][kk + 1];
      v2f b0; b0.x = Bs[buf][kk][mrow];      b0.y = Bs[buf][kk + 1][mrow];
      v2f b1; b1.x = Bs[buf][kk][mrow + 16]; b1.y = Bs[buf][kk + 1][mrow + 16];
      c0 = wmma_f32(a, b0, c0);   // N-tile 0 (cols 0..15)
      c1 = wmma_f32(a, b1, c1);   // N-tile 1 (cols 16..31)
    }
    wait_async0();     // next buffer's async copies landed
    __syncthreads();   // all waves done reading `buf` before it is refilled
  }

  // cross-wave reduction of K-partials, then store
  #pragma unroll
  for (int t = 0; t < 2; ++t) {
    const v8f c = t ? c1 : c0;
    #pragma unroll
    for (int r = 0; r < 8; ++r) red[wave][r * 32 + lane] = c[r];
    __syncthreads();
    float s = 0.f;
    #pragma unroll
    for (int w = 0; w < 8; ++w) s += red[w][tid];
    const int r  = tid >> 5;
    const int ln = tid & 31;
    const long row = rowBase + r + ((ln >> 4) << 3);   // C layout: M = r / r+8
    const int  col = t * 16 + (ln & 15);
    C[row * DPAD + col] = s;
    __syncthreads();
  }
}

// ---------------------------------------------------------------------------
// x = agg @ ggn_w.T + ggn_b ; h = GRUCell(x, h).  Thread-per-atom, weights in LDS.
// ---------------------------------------------------------------------------
__global__ __launch_bounds__(256) void k_ggn_gru(const float* __restrict__ agg,
                                                 float* __restrict__ h,
                                                 const float* __restrict__ ggn_w,
                                                 const float* __restrict__ ggn_b,
                                                 const float* __restrict__ wih,
                                                 const float* __restrict__ whh,
                                                 const float* __restrict__ bih,
                                                 const float* __restrict__ bhh) {
  __shared__ float sWg[625], sBg[25], sWih[1875], sWhh[1875], sBih[75], sBhh[75];
  const int tid = threadIdx.x;
  for (int t = tid; t < 625;  t += 256) sWg[t]  = ggn_w[t];
  for (int t = tid; t < 25;   t += 256) sBg[t]  = ggn_b[t];
  for (int t = tid; t < 1875; t += 256) { sWih[t] = wih[t]; sWhh[t] = whh[t]; }
  for (int t = tid; t < 75;   t += 256) { sBih[t] = bih[t]; sBhh[t] = bhh[t]; }
  __syncthreads();

  const int atom = blockIdx.x * 256 + tid;
  if (atom >= N_ATOMS) return;

  float ar[D_IN], hr[D_IN], x[D_IN], hn[D_IN];
  #pragma unroll
  for (int d = 0; d < D_IN; ++d) {
    ar[d] = agg[atom * DPAD + d];
    hr[d] = h[atom * DPAD + d];
  }
  #pragma unroll 1
  for (int d = 0; d < D_IN; ++d) {
    float s = sBg[d];
    for (int k = 0; k < D_IN; ++k) s += ar[k] * sWg[d * D_IN + k];
    x[d] = s;
  }
  #pragma unroll 1
  for (int d = 0; d < D_IN; ++d) {
    float ir = sBih[d],      hrv = sBhh[d];
    float iz = sBih[25 + d], hz  = sBhh[25 + d];
    float in_ = sBih[50 + d], hnn = sBhh[50 + d];
    for (int k = 0; k < D_IN; ++k) {
      ir  += x[k]  * sWih[d * D_IN + k];
      hrv += hr[k] * sWhh[d * D_IN + k];
      iz  += x[k]  * sWih[(25 + d) * D_IN + k];
      hz  += hr[k] * sWhh[(25 + d) * D_IN + k];
      in_ += x[k]  * sWih[(50 + d) * D_IN + k];
      hnn += hr[k] * sWhh[(50 + d) * D_IN + k];
    }
    const float r = 1.f / (1.f + __expf(-(ir + hrv)));
    const float z = 1.f / (1.f + __expf(-(iz + hz)));
    const float n = tanhf(in_ + r * hnn);
    hn[d] = (1.f - z) * n + z * hr[d];
  }
  #pragma unroll
  for (int d = 0; d < D_IN; ++d) h[atom * DPAD + d] = hn[d];
}

// ---------------------------------------------------------------------------
// Pair FFN: one wave = 16 pairs.  mean -> relu(W1)+b1 -> sigmoid(W2)+b2.
// Both GEMMs via V_WMMA_F32_16X16X4_F32 (K padded to 32 with zeros).
// ---------------------------------------------------------------------------
__global__ __launch_bounds__(256) void k_pairs_ffn(const float* __restrict__ hf,
                                                   const float* __restrict__ w1,
                                                   const float* __restrict__ b1,
                                                   const float* __restrict__ w2,
                                                   const float* __restrict__ b2,
                                                   float* __restrict__ out,
                                                   long P) {
  __shared__ float xs[8][16][36];   // per-wave relu output, C-layout -> A-layout transpose
  const int tid   = threadIdx.x;
  const int wave  = tid >> 5;
  const int lane  = tid & 31;
  const int m     = lane & 15;
  const int khalf = (lane >> 4) << 1;
  const long pbase = ((long)blockIdx.x * 8 + wave) * 16;

  long p = pbase + m;
  if (p >= P) p = P - 1;                       // clamp: keeps EXEC all-ones for WMMA
  // invert row-major upper-triangular index: i = floor((2M+1 - sqrt((2M+1)^2-8p))/2)
  const double tq = 2.0 * (double)N_MEMBERS + 1.0;
  long i = (long)((tq - __builtin_sqrt(tq * tq - 8.0 * (double)p)) * 0.5);
  if (i < 0) i = 0;
  if (i >= N_MEMBERS) i = N_MEMBERS - 1;
  auto offs = [](long r) { return r * (2L * N_MEMBERS - r + 1) / 2; };
  while (offs(i + 1) <= p) ++i;
  while (offs(i) > p) --i;
  const long j = i + (p - offs(i));
  const float* hi = hf + i * DPAD;
  const float* hj = hf + j * DPAD;

  // stage 1: [16x32] @ W1^T (zero-padded to 32x32)
  v8f c0 = {}; v8f c1 = {};
  #pragma unroll
  for (int k0 = 0; k0 < 32; k0 += 4) {
    const int kk = k0 + khalf;
    v2f a; a.x = 0.5f * (hi[kk] + hj[kk]); a.y = 0.5f * (hi[kk + 1] + hj[kk + 1]);
    const int n1 = m + 16;
    v2f bb0, bb1;
    bb0.x = (kk     < 25) ? w1[m * 25 + kk]     : 0.f;
    bb0.y = (kk + 1 < 25) ? w1[m * 25 + kk + 1] : 0.f;
    bb1.x = (n1 < 25 && kk     < 25) ? w1[n1 * 25 + kk]     : 0.f;
    bb1.y = (n1 < 25 && kk + 1 < 25) ? w1[n1 * 25 + kk + 1] : 0.f;
    c0 = wmma_f32(a, bb0, c0);
    c1 = wmma_f32(a, bb1, c1);
  }
  // bias + relu, write to LDS in [pair][feature] order (C-layout -> A-layout)
  #pragma unroll
  for (int r = 0; r < 8; ++r) {
    const int row = r + ((lane >> 4) << 3);
    const int n0 = lane & 15, n1 = n0 + 16;
    xs[wave][row][n0] = fmaxf(c0[r] + b1[n0], 0.f);
    xs[wave][row][n1] = (n1 < 25) ? fmaxf(c1[r] + b1[n1], 0.f) : 0.f;
  }
  // stage 2: [16x32] @ W2^T (12 valid output cols)
  v8f c2 = {};
  #pragma unroll
  for (int k0 = 0; k0 < 32; k0 += 4) {
    const int kk = k0 + khalf;
    v2f a; a.x = xs[wave][m][kk]; a.y = xs[wave][m][kk + 1];
    v2f b;
    b.x = (m < 12 && kk     < 25) ? w2[m * 25 + kk]     : 0.f;
    b.y = (m < 12 && kk + 1 < 25) ? w2[m * 25 + kk + 1] : 0.f;
    c2 = wmma_f32(a, b, c2);
  }
  #pragma unroll
  for (int r = 0; r < 8; ++r) {
    const int row = r + ((lane >> 4) << 3);
    const int n   = lane & 15;
    const long pp = pbase + row;
    if (n < 12 && pp < P) {
      const float v = c2[r] + b2[n];
      out[pp * 12 + n] = 1.f / (1.f + __expf(-v));
    }
  }
}

// ---------------------------------------------------------------------------
extern "C" void kernel_launch(void* const* d_in, const int* in_sizes, int n_in,
                              void* d_out, int out_size, void* d_ws, size_t ws_size,
                              hipStream_t stream) {
  (void)in_sizes; (void)n_in; (void)out_size; (void)ws_size;
  const float* adjacency  = (const float*)d_in[0];
  const int*   atoms      = (const int*)  d_in[1];
  const float* membership = (const float*)d_in[2];
  const float* embed      = (const float*)d_in[3];
  const float* ggn_w = (const float*)d_in[4];
  const float* ggn_b = (const float*)d_in[5];
  const float* wih   = (const float*)d_in[6];
  const float* whh   = (const float*)d_in[7];
  const float* bih   = (const float*)d_in[8];
  const float* bhh   = (const float*)d_in[9];
  const float* w1    = (const float*)d_in[10];
  const float* b1    = (const float*)d_in[11];
  const float* w2    = (const float*)d_in[12];
  const float* b2    = (const float*)d_in[13];
  float* out = (float*)d_out;

  char* ws = (char*)d_ws;
  float* h   = (float*)ws;                                        // 8000*32 f32
  float* agg = (float*)(ws + (size_t)N_ATOMS * DPAD * 4);         // 8000*32 f32
  float* hf  = (float*)(ws + (size_t)2 * N_ATOMS * DPAD * 4);     // 2000*32 f32

  k_gather<<<(N_ATOMS * DPAD + 255) / 256, 256, 0, stream>>>(atoms, embed, h);

  for (int it = 0; it < 4; ++it) {
    k_gemm_wmma<<<N_ATOMS / 16, 256, 0, stream>>>(adjacency, h, agg, N_ATOMS);
    k_ggn_gru<<<(N_ATOMS + 255) / 256, 256, 0, stream>>>(agg, h, ggn_w, ggn_b,
                                                         wih, whh, bih, bhh);
  }
  k_gemm_wmma<<<N_MEMBERS / 16, 256, 0, stream>>>(membership, h, hf, N_ATOMS);

  const long P = (long)N_MEMBERS * (N_MEMBERS + 1) / 2;   // 2,001,000
  const long waves = (P + 15) / 16;
  const int  blocks = (int)((waves + 7) / 8);
  k_pairs_ffn<<<blocks, 256, 0, stream>>>(hf, w1, b1, w2, b2, out, P);
}